// Block_graph_74560632259322
// MI455X (gfx1250) — compile-verified
//
#include <hip/hip_runtime.h>
#include <math.h>

#define N_NODES 4096
#define H_DIM   768
#define K_NBR   32
#define EPS_MSG 1e-7f
#define LN_EPS  1e-5f

typedef __attribute__((ext_vector_type(2))) float v2f;
typedef __attribute__((ext_vector_type(8))) float v8f;

#define WMMA_F32(A, B, C) \
  __builtin_amdgcn_wmma_f32_16x16x4_f32(false, (A), false, (B), (short)0, (C), false, false)

// ---------------------------------------------------------------------------
// GENConv softmax aggregation: pre[i,c] = softmax_agg(relu(x[src])+eps) + x[i,c]
// One block (256 threads) per node; online softmax over the K=32 neighbors.
// ---------------------------------------------------------------------------
__global__ __launch_bounds__(256) void genconv_agg(
    const float* __restrict__ x, const int* __restrict__ src,
    const float* __restrict__ tptr, float* __restrict__ pre) {
  __shared__ int s_src[K_NBR];
  const int node = blockIdx.x;
  const int tid  = threadIdx.x;
  if (tid < K_NBR) s_src[tid] = src[node * K_NBR + tid];
  __syncthreads();
  const float t = tptr[0];
  #pragma unroll
  for (int j = 0; j < 3; ++j) {
    const int c = tid + j * 256;
    float m = -INFINITY, num = 0.f, den = 0.f;
    #pragma unroll 8
    for (int k = 0; k < K_NBR; ++k) {
      const float v   = x[(size_t)s_src[k] * H_DIM + c];
      const float msg = fmaxf(v, 0.f) + EPS_MSG;
      const float lg  = t * msg;
      const float mn  = fmaxf(m, lg);
      const float sc  = __expf(m - mn);   // 0 on first iter (m = -inf)
      const float w   = __expf(lg - mn);
      num = num * sc + w * msg;
      den = den * sc + w;
      m = mn;
    }
    pre[(size_t)node * H_DIM + c] = num / den + x[(size_t)node * H_DIM + c];
  }
}

// ---------------------------------------------------------------------------
// WMMA f32 GEMM: C[M x 768] = A[M x KTOT] * W[768 x KTOT]^T + bias
// CAT=true: A is the concat [A0 | A1], each with row stride 768.
// Wave computes a 64-row x 32-col tile (4 row-tiles x 2 col-tiles, 8 v8f
// accumulators) so each A fragment feeds two WMMAs: per K-step (4 k) we load
// 4 A + 2 B b64 fragments and issue 8 v_wmma (0.75 loads/WMMA).
// Block = 8 waves = 256 cols. Grid: (M/64, 768/256).
// VGPR layout per ISA: A/B K = vgpr + 2*(lane/16), A-row / B-col = lane%16;
// C/D row = vgpr + 8*(lane/16), col = lane%16.
// ---------------------------------------------------------------------------
template <int KTOT, bool CAT>
__global__ __launch_bounds__(256) void gemm_wmma(
    const float* __restrict__ A0, const float* __restrict__ A1,
    const float* __restrict__ W,  const float* __restrict__ bias,
    float* __restrict__ C) {
  const int lane = threadIdx.x & 31;
  const int wave = threadIdx.x >> 5;
  const int m0   = blockIdx.x * 64;
  const int n0   = blockIdx.y * 256 + wave * 32;  // col tiles at n0 and n0+16
  const int rc   = lane & 15;   // A row in tile / B,C column
  const int half = lane >> 4;

  const float bj0 = bias[n0 + rc];
  const float bj1 = bias[n0 + 16 + rc];
  v8f c00, c10, c20, c30, c01, c11, c21, c31;
  #pragma unroll
  for (int r = 0; r < 8; ++r) {
    c00[r] = bj0; c10[r] = bj0; c20[r] = bj0; c30[r] = bj0;
    c01[r] = bj1; c11[r] = bj1; c21[r] = bj1; c31[r] = bj1;
  }

  const float* Wp0 = W + (size_t)(n0 + rc) * KTOT + 2 * half;
  const float* Wp1 = Wp0 + (size_t)16 * KTOT;

  #pragma unroll 4
  for (int kk = 0; kk < KTOT; kk += 4) {
    const v2f b0 = *(const v2f*)(Wp0 + kk);
    const v2f b1 = *(const v2f*)(Wp1 + kk);
    const float* Asrc;
    int kloc;
    if (CAT) {
      Asrc = (kk < H_DIM) ? A0 : A1;      // uniform branch; k-quads don't straddle 768
      kloc = (kk < H_DIM) ? kk : (kk - H_DIM);
    } else {
      Asrc = A0;
      kloc = kk;
    }
    const float* Ap = Asrc + (size_t)(m0 + rc) * H_DIM + kloc + 2 * half;
    const v2f a0 = *(const v2f*)(Ap);
    const v2f a1 = *(const v2f*)(Ap + 16 * H_DIM);
    const v2f a2 = *(const v2f*)(Ap + 32 * H_DIM);
    const v2f a3 = *(const v2f*)(Ap + 48 * H_DIM);
    c00 = WMMA_F32(a0, b0, c00);
    c01 = WMMA_F32(a0, b1, c01);
    c10 = WMMA_F32(a1, b0, c10);
    c11 = WMMA_F32(a1, b1, c11);
    c20 = WMMA_F32(a2, b0, c20);
    c21 = WMMA_F32(a2, b1, c21);
    c30 = WMMA_F32(a3, b0, c30);
    c31 = WMMA_F32(a3, b1, c31);
  }

  #pragma unroll
  for (int r = 0; r < 8; ++r) {
    const int row = r + 8 * half;
    float* Cr0 = C + (size_t)(m0 + row) * H_DIM + n0 + rc;
    Cr0[ 0 * H_DIM +  0] = c00[r];
    Cr0[ 0 * H_DIM + 16] = c01[r];
    Cr0[16 * H_DIM +  0] = c10[r];
    Cr0[16 * H_DIM + 16] = c11[r];
    Cr0[32 * H_DIM +  0] = c20[r];
    Cr0[32 * H_DIM + 16] = c21[r];
    Cr0[48 * H_DIM +  0] = c30[r];
    Cr0[48 * H_DIM + 16] = c31[r];
  }
}

// ---------------------------------------------------------------------------
// x2 = x1 + relu(LayerNorm(h2)); in-place on h2. Block per node row.
// ---------------------------------------------------------------------------
__global__ __launch_bounds__(256) void ln_relu_add(
    float* __restrict__ h2, const float* __restrict__ x1,
    const float* __restrict__ g, const float* __restrict__ b) {
  __shared__ float sred[256];
  const int node = blockIdx.x, tid = threadIdx.x;
  const size_t base = (size_t)node * H_DIM;
  const float v0 = h2[base + tid], v1 = h2[base + tid + 256], v2 = h2[base + tid + 512];

  sred[tid] = v0 + v1 + v2;
  __syncthreads();
  for (int off = 128; off > 0; off >>= 1) {
    if (tid < off) sred[tid] += sred[tid + off];
    __syncthreads();
  }
  const float mu = sred[0] * (1.f / H_DIM);
  __syncthreads();

  sred[tid] = v0 * v0 + v1 * v1 + v2 * v2;
  __syncthreads();
  for (int off = 128; off > 0; off >>= 1) {
    if (tid < off) sred[tid] += sred[tid + off];
    __syncthreads();
  }
  const float var = sred[0] * (1.f / H_DIM) - mu * mu;
  const float rs  = rsqrtf(var + LN_EPS);

  #pragma unroll
  for (int j = 0; j < 3; ++j) {
    const int c = tid + j * 256;
    const float v = (j == 0) ? v0 : (j == 1) ? v1 : v2;
    const float ln = (v - mu) * rs * g[c] + b[c];
    h2[base + c] = x1[base + c] + fmaxf(ln, 0.f);
  }
}

// ---------------------------------------------------------------------------
// Column mean of g over N rows: zero + partial atomic sums + finalize.
// ---------------------------------------------------------------------------
__global__ void zero_mean(float* __restrict__ mean) {
  mean[blockIdx.x * 256 + threadIdx.x] = 0.f;
}

__global__ __launch_bounds__(256) void colsum(
    const float* __restrict__ g, float* __restrict__ mean) {
  const int c  = blockIdx.x * 256 + threadIdx.x;
  const int r0 = blockIdx.y * 128;
  float s = 0.f;
  #pragma unroll 4
  for (int r = r0; r < r0 + 128; ++r) s += g[(size_t)r * H_DIM + c];
  atomicAdd(mean + c, s);
}

__global__ __launch_bounds__(256) void finalize(
    const float* __restrict__ x, const float* __restrict__ mean,
    float* __restrict__ out) {
  const size_t idx = (size_t)blockIdx.x * 256 + threadIdx.x;
  float v = x[idx];
  if (idx < H_DIM) v += mean[idx] * (1.f / N_NODES);
  out[idx] = v;
}

// ---------------------------------------------------------------------------
extern "C" void kernel_launch(void* const* d_in, const int* in_sizes, int n_in,
                              void* d_out, int out_size, void* d_ws, size_t ws_size,
                              hipStream_t stream) {
  const float* x    = (const float*)d_in[0];
  const float* W1   = (const float*)d_in[1];
  const float* b1   = (const float*)d_in[2];
  const float* t1   = (const float*)d_in[3];
  const float* W2   = (const float*)d_in[4];
  const float* b2   = (const float*)d_in[5];
  const float* t2   = (const float*)d_in[6];
  const float* ln_g = (const float*)d_in[7];
  const float* ln_b = (const float*)d_in[8];
  const float* Wc   = (const float*)d_in[9];
  const float* bc   = (const float*)d_in[10];
  const int*   src  = (const int*)d_in[11];   // first E = N*K entries
  float* out = (float*)d_out;

  const size_t NH = (size_t)N_NODES * H_DIM;
  float* buf_pre = (float*)d_ws;              // pre-activations, reused for g
  float* buf_x1  = buf_pre + NH;
  float* buf_x2  = buf_x1 + NH;               // h2 -> x2 in place
  float* mean    = buf_x2 + NH;

  const dim3 gemm_grid(N_NODES / 64, H_DIM / 256);

  // Layer 1: aggregate + GEMM
  genconv_agg<<<N_NODES, 256, 0, stream>>>(x, src, t1, buf_pre);
  gemm_wmma<H_DIM, false><<<gemm_grid, 256, 0, stream>>>(buf_pre, nullptr, W1, b1, buf_x1);
  // Layer 2: aggregate + GEMM + LN/ReLU/residual
  genconv_agg<<<N_NODES, 256, 0, stream>>>(buf_x1, src, t2, buf_pre);
  gemm_wmma<H_DIM, false><<<gemm_grid, 256, 0, stream>>>(buf_pre, nullptr, W2, b2, buf_x2);
  ln_relu_add<<<N_NODES, 256, 0, stream>>>(buf_x2, buf_x1, ln_g, ln_b);
  // Concat GEMM: g = [x1|x2] @ Wc^T + bc  -> buf_pre
  gemm_wmma<2 * H_DIM, true><<<gemm_grid, 256, 0, stream>>>(buf_x1, buf_x2, Wc, bc, buf_pre);
  // mean over nodes + residual onto CLS row
  zero_mean<<<H_DIM / 256, 256, 0, stream>>>(mean);
  colsum<<<dim3(H_DIM / 256, N_NODES / 128), 256, 0, stream>>>(buf_pre, mean);
  finalize<<<(unsigned)(NH / 256), 256, 0, stream>>>(x, mean, out);
}